// GraphSAGENet_82609400971648
// MI455X (gfx1250) — compile-verified
//
#include <hip/hip_runtime.h>
#include <hip/hip_bf16.h>
#include <math.h>

typedef __attribute__((ext_vector_type(2))) float v2f;
typedef __attribute__((ext_vector_type(8))) float v8f;

#define N_NODES    50000
#define N_EDGES    400000
#define IN_DIM     512
#define HID        512
#define HID2       256
#define NUM_GRAPHS 500

// ---------------- utility ----------------
__global__ void zero_i32(int* p, int n)   { int i = blockIdx.x*blockDim.x + threadIdx.x; if (i < n) p[i] = 0; }
__global__ void zero_f32(float* p, int n) { int i = blockIdx.x*blockDim.x + threadIdx.x; if (i < n) p[i] = 0.f; }

// ---------------- CSR build (by destination) ----------------
__global__ void hist_kernel(const long long* __restrict__ dst, int* __restrict__ deg, int n) {
    int e = blockIdx.x*blockDim.x + threadIdx.x;
    if (e < n) atomicAdd(&deg[(int)dst[e]], 1);
}

// single-block exclusive scan over 50000 ints (Hillis-Steele per 1024-chunk + carry)
__global__ void scan_kernel(const int* __restrict__ deg, int* __restrict__ rowstart, int n) {
    __shared__ int buf[1024];
    __shared__ int carry;
    int tid = threadIdx.x;
    if (tid == 0) carry = 0;
    __syncthreads();
    for (int base = 0; base < n; base += 1024) {
        int v = (base + tid < n) ? deg[base + tid] : 0;
        buf[tid] = v;
        __syncthreads();
        for (int off = 1; off < 1024; off <<= 1) {
            int t = (tid >= off) ? buf[tid - off] : 0;
            __syncthreads();
            buf[tid] += t;
            __syncthreads();
        }
        if (base + tid < n) rowstart[base + tid] = carry + buf[tid] - v;  // exclusive
        int total = buf[1023];
        __syncthreads();
        if (tid == 0) carry += total;
        __syncthreads();
    }
    if (tid == 0) rowstart[n] = carry;
}

__global__ void scatter_kernel(const long long* __restrict__ src, const long long* __restrict__ dst,
                               const int* __restrict__ rowstart, int* __restrict__ cursor,
                               int* __restrict__ csr, int n) {
    int e = blockIdx.x*blockDim.x + threadIdx.x;
    if (e < n) {
        int d   = (int)dst[e];
        int pos = atomicAdd(&cursor[d], 1);
        csr[rowstart[d] + pos] = (int)src[e];
    }
}

// ---------------- fp32 WMMA GEMM: C[M,N] = A[M,K] * W[N,K]^T ----------------
// block = 256 threads (8 waves), handles a 32-row M super-strip (two 16-row tiles).
// A super-strip staged into LDS with GLOBAL_LOAD_ASYNC_TO_LDS_B128 (ASYNCcnt path,
// no VGPR round-trip), padded stride -> conflict-free ds_load_b64 fragments.
// Each wave owns NT 16-col N tiles; B fragments double-buffered and shared by the
// two M tiles (2*NT WMMAs per k-step, NT global b64 loads).
// f32 16x16x4 layouts: A lane L -> A[M=L%16][K=2*(L/16)+v]; B mirrors with N=L%16.
template <int NT>
__global__ __launch_bounds__(256) void gemm_wmma_kernel(const float* __restrict__ A,
                                                        const float* __restrict__ W,
                                                        float* __restrict__ C,
                                                        int M, int N, int K) {
    constexpr int KMAX = 512;
    constexpr int PAD  = 4;
    __shared__ float As[32 * (KMAX + PAD)];
    const int LSTR = K + PAD;

    const int tid  = threadIdx.x;
    const int wave = tid >> 5;
    const int lane = tid & 31;
    const int half = lane >> 4;
    const int l16  = lane & 15;
    const int m0   = blockIdx.x * 32;
    const int n0   = wave * (NT * 16);
    const int koff = 2 * half;
    const bool has2 = (m0 + 16) < M;

    // ---- async stage of up to 32 rows x K into LDS (b128 per lane, no VGPR copy) ----
    {
        const int rows   = has2 ? 32 : 16;
        const int total4 = rows * (K >> 2);          // uniform trip count across threads
        for (int i = tid; i < total4; i += 256) {
            int j   = i << 2;
            int row = j / K;
            int col = j - row * K;
            const float* g = A + (size_t)(m0 + row) * K + col;
            unsigned lds = (unsigned)(size_t)(&As[row * LSTR + col]);  // wave-relative LDS byte addr
            asm volatile("global_load_async_to_lds_b128 %0, %1, off"
                         :: "v"(lds), "v"(g) : "memory");
        }
        asm volatile("s_wait_asynccnt 0x0" ::: "memory");
    }
    __syncthreads();

    // 32-bit element offsets into W (all buffers < 4GB)
    unsigned wOff[NT];
#pragma unroll
    for (int t = 0; t < NT; ++t) wOff[t] = (unsigned)(n0 + 16*t + l16) * (unsigned)K + (unsigned)koff;

    v8f acc1[NT], acc2[NT];
#pragma unroll
    for (int t = 0; t < NT; ++t) {
        acc1[t] = (v8f){0.f,0.f,0.f,0.f,0.f,0.f,0.f,0.f};
        acc2[t] = (v8f){0.f,0.f,0.f,0.f,0.f,0.f,0.f,0.f};
    }

    // ---- software-pipelined k-loop (double-buffered A/B fragments) ----
    const unsigned aIdx1 = (unsigned)(l16 * LSTR + koff);
    const unsigned aIdx2 = (unsigned)((16 + l16) * LSTR + koff);
    v2f a1c = *(const v2f*)&As[aIdx1];
    v2f a2c = *(const v2f*)&As[aIdx2];
    v2f bc[NT];
#pragma unroll
    for (int t = 0; t < NT; ++t) bc[t] = *(const v2f*)(W + wOff[t]);

    for (int k0 = 0; k0 < K; k0 += 4) {
        const int kp = (k0 + 4 < K) ? (k0 + 4) : k0;   // last iter: benign reload
        v2f a1n = *(const v2f*)&As[aIdx1 + kp];
        v2f a2n = *(const v2f*)&As[aIdx2 + kp];
        v2f bn[NT];
#pragma unroll
        for (int t = 0; t < NT; ++t) bn[t] = *(const v2f*)(W + wOff[t] + kp);

#pragma unroll
        for (int t = 0; t < NT; ++t)
            acc1[t] = __builtin_amdgcn_wmma_f32_16x16x4_f32(false, a1c, false, bc[t],
                                                            (short)0, acc1[t], false, false);
#pragma unroll
        for (int t = 0; t < NT; ++t)
            acc2[t] = __builtin_amdgcn_wmma_f32_16x16x4_f32(false, a2c, false, bc[t],
                                                            (short)0, acc2[t], false, false);
        a1c = a1n; a2c = a2n;
#pragma unroll
        for (int t = 0; t < NT; ++t) bc[t] = bn[t];
    }

    // ---- store: VGPR r -> C[m + r + 8*half][n0 + 16*t + l16] ----
#pragma unroll
    for (int t = 0; t < NT; ++t) {
#pragma unroll
        for (int r = 0; r < 8; ++r) {
            C[(unsigned)(m0 + r + 8*half) * (unsigned)N + (unsigned)(n0 + 16*t + l16)] = acc1[t][r];
        }
    }
    if (has2) {
#pragma unroll
        for (int t = 0; t < NT; ++t) {
#pragma unroll
            for (int r = 0; r < 8; ++r) {
                C[(unsigned)(m0 + 16 + r + 8*half) * (unsigned)N + (unsigned)(n0 + 16*t + l16)] = acc2[t][r];
            }
        }
    }
}

// ---------------- fused mean-aggregate + bias + root + ReLU (float2/lane) ----------------
// out[n] = relu( (1/max(deg,1)) * sum_{e in CSR(n)} Zl[src_e] + bias + Zr[n] )
// launch with blockDim = D/2
__global__ void sage_combine_kernel(const float* __restrict__ Zl, const float* __restrict__ Zr,
                                    const float* __restrict__ bias,
                                    const int* __restrict__ rowstart, const int* __restrict__ csr,
                                    float* __restrict__ out, int D) {
    const int node = blockIdx.x;
    const int e0 = rowstart[node], e1 = rowstart[node + 1];
    const float inv = 1.0f / fmaxf((float)(e1 - e0), 1.0f);
    const unsigned f2 = threadIdx.x * 2u;
    float ax = 0.f, ay = 0.f;
    for (int j = e0; j < e1; ++j) {
        const unsigned s = (unsigned)csr[j];
        v2f z = *(const v2f*)(Zl + (size_t)s * D + f2);
        ax += z.x; ay += z.y;
    }
    v2f r = *(const v2f*)(Zr + (size_t)node * D + f2);
    v2f o;
    o.x = fmaxf(ax * inv + bias[f2]     + r.x, 0.f);
    o.y = fmaxf(ay * inv + bias[f2 + 1] + r.y, 0.f);
    *(v2f*)(out + (size_t)node * D + f2) = o;
}

// ---------------- global mean pool (atomic scatter into 512KB L2-resident buffer) ----------------
__global__ void pool_kernel(const float* __restrict__ h, const long long* __restrict__ batch,
                            float* __restrict__ pooled, float* __restrict__ counts) {
    int node = blockIdx.x;
    int g = (int)batch[node];
    int f = threadIdx.x;
    atomicAdd(&pooled[(size_t)g * HID2 + f], h[(size_t)node * HID2 + f]);
    if (f == 0) atomicAdd(&counts[g], 1.0f);
}

// ---------------- head: fc + log_softmax (one wave per graph) ----------------
__global__ void head_kernel(const float* __restrict__ pooled, const float* __restrict__ counts,
                            const float* __restrict__ Wfc, const float* __restrict__ bfc,
                            float* __restrict__ out) {
    int g = blockIdx.x;
    int lane = threadIdx.x;
    float cnt = fmaxf(counts[g], 1.0f);
    float s0 = 0.f, s1 = 0.f;
    for (int f = lane; f < HID2; f += 32) {
        float p = pooled[(size_t)g * HID2 + f] / cnt;
        s0 += p * Wfc[f];
        s1 += p * Wfc[HID2 + f];
    }
    for (int off = 16; off > 0; off >>= 1) {
        s0 += __shfl_xor(s0, off, 32);
        s1 += __shfl_xor(s1, off, 32);
    }
    if (lane == 0) {
        float l0 = s0 + bfc[0], l1 = s1 + bfc[1];
        float m = fmaxf(l0, l1);
        float lse = m + logf(expf(l0 - m) + expf(l1 - m));
        out[(size_t)g * 2 + 0] = l0 - lse;
        out[(size_t)g * 2 + 1] = l1 - lse;
    }
}

extern "C" void kernel_launch(void* const* d_in, const int* in_sizes, int n_in,
                              void* d_out, int out_size, void* d_ws, size_t ws_size,
                              hipStream_t stream) {
    const float*     x    = (const float*)d_in[0];
    const long long* ei   = (const long long*)d_in[1];   // int64 [2, E]
    const long long* batch= (const long long*)d_in[2];   // int64 [N]
    const float*     W1l  = (const float*)d_in[3];
    const float*     b1   = (const float*)d_in[4];
    const float*     W1r  = (const float*)d_in[5];
    const float*     W2l  = (const float*)d_in[6];
    const float*     b2   = (const float*)d_in[7];
    const float*     W2r  = (const float*)d_in[8];
    const float*     Wfc  = (const float*)d_in[9];
    const float*     bfc  = (const float*)d_in[10];
    float* out = (float*)d_out;

    const long long* e_src = ei;
    const long long* e_dst = ei + N_EDGES;

    // workspace layout
    char* ws = (char*)d_ws;
    const size_t BUF = (size_t)N_NODES * HID * sizeof(float);     // 102,400,000 B
    float* Z1l = (float*)(ws);                                    // [N, 512]
    float* Z1r = (float*)(ws + BUF);                              // [N, 512]
    float* h1  = (float*)(ws + 2*BUF);                            // [N, 512]
    float* Z2l = Z1l;                                             // reuse: [N, 256]
    float* Z2r = Z1l + (size_t)N_NODES * HID2;                    // reuse: [N, 256]
    float* h2  = Z1r;                                             // reuse: [N, 256]
    int* ints      = (int*)(ws + 3*BUF);
    int* deg       = ints;                       // [N]
    int* cursor    = deg + N_NODES;              // [N]
    int* rowstart  = cursor + N_NODES;           // [N+1]
    int* csr       = rowstart + N_NODES + 1;     // [E]
    float* pooled  = (float*)(csr + N_EDGES);    // [G, 256]
    float* counts  = pooled + (size_t)NUM_GRAPHS * HID2;  // [G]

    // 1) init + CSR build
    zero_i32<<<(N_NODES + 255)/256, 256, 0, stream>>>(deg, N_NODES);
    zero_i32<<<(N_NODES + 255)/256, 256, 0, stream>>>(cursor, N_NODES);
    zero_f32<<<(NUM_GRAPHS*HID2 + 255)/256, 256, 0, stream>>>(pooled, NUM_GRAPHS*HID2);
    zero_f32<<<(NUM_GRAPHS + 255)/256, 256, 0, stream>>>(counts, NUM_GRAPHS);
    hist_kernel<<<(N_EDGES + 255)/256, 256, 0, stream>>>(e_dst, deg, N_EDGES);
    scan_kernel<<<1, 1024, 0, stream>>>(deg, rowstart, N_NODES);
    scatter_kernel<<<(N_EDGES + 255)/256, 256, 0, stream>>>(e_src, e_dst, rowstart, cursor, csr, N_EDGES);

    const int MBLOCKS = (N_NODES + 31) / 32;   // 1563 (last block: single strip)

    // 2) layer 1: Z = x @ W^T (WMMA fp32), then aggregate+bias+root+relu
    gemm_wmma_kernel<4><<<MBLOCKS, 256, 0, stream>>>(x, W1l, Z1l, N_NODES, HID, IN_DIM);
    gemm_wmma_kernel<4><<<MBLOCKS, 256, 0, stream>>>(x, W1r, Z1r, N_NODES, HID, IN_DIM);
    sage_combine_kernel<<<N_NODES, HID/2, 0, stream>>>(Z1l, Z1r, b1, rowstart, csr, h1, HID);

    // 3) layer 2
    gemm_wmma_kernel<2><<<MBLOCKS, 256, 0, stream>>>(h1, W2l, Z2l, N_NODES, HID2, HID);
    gemm_wmma_kernel<2><<<MBLOCKS, 256, 0, stream>>>(h1, W2r, Z2r, N_NODES, HID2, HID);
    sage_combine_kernel<<<N_NODES, HID2/2, 0, stream>>>(Z2l, Z2r, b2, rowstart, csr, h2, HID2);

    // 4) pool + head
    pool_kernel<<<N_NODES, HID2, 0, stream>>>(h2, batch, pooled, counts);
    head_kernel<<<NUM_GRAPHS, 32, 0, stream>>>(pooled, counts, Wfc, bfc, out);
}